// GATModel_23124103922204
// MI455X (gfx1250) — compile-verified
//
#include <hip/hip_runtime.h>
#include <math.h>

#define N_NODES 50000
#define E_EDGES 800000
#define E_TOTAL (N_NODES + E_EDGES)
#define G_GRAPHS 64
#define NEG_SLOPE 0.2f

typedef __attribute__((ext_vector_type(2))) float v2f;
typedef __attribute__((ext_vector_type(8))) float v8f;

// ---------------------------------------------------------------------------
// Strip-mined WMMA fp32 GEMM: H[nrows, DO] = X[nrows, K] @ W[K, DO]
// Each wave computes a 16 x (16*NT) output strip with V_WMMA_F32_16X16X4_F32.
// The A fragment (16x4 chunk of X) is loaded ONCE per K-step and reused across
// NT column tiles; B fragments come from the tiny L1-resident W matrix.
// nrows = 50000 = 3125*16, K/DO multiples of 16 -> no remainder handling.
// A 16x4: lanes 0-15 -> M=lane, VGPR0/1 = K+0/K+1; lanes 16-31 -> K+2/K+3
// B 4x16: lanes 0-15 -> N=lane, VGPR0/1 = K+0/K+1; lanes 16-31 -> K+2/K+3
// C/D:    lane&15 = N, VGPR r -> M = r + (lane>>4)*8
// ---------------------------------------------------------------------------
template<int K, int DO, int NT>
__global__ __launch_bounds__(256) void gemm_wmma_f32(const float* __restrict__ X,
                                                     const float* __restrict__ Wm,
                                                     float* __restrict__ H, int nrows)
{
    const int lane = threadIdx.x & 31;
    const int wave = threadIdx.x >> 5;
    constexpr int NSTRIP = DO / (16 * NT);        // strips per row-tile
    const int sid   = blockIdx.x * 8 + wave;      // global strip id (wave-uniform)
    const int total = (nrows >> 4) * NSTRIP;
    if (sid >= total) return;                     // wave-uniform: EXEC stays all-1s
    const int tm   = (sid / NSTRIP) << 4;
    const int tn0  = (sid % NSTRIP) * (16 * NT);
    const int m    = lane & 15;
    const int half = lane >> 4;

    v8f acc[NT];
#pragma unroll
    for (int t = 0; t < NT; ++t) acc[t] = (v8f){};

    const float* xrow = X + (size_t)(tm + m) * K;
    for (int k0 = 0; k0 < K; k0 += 4) {
        const int ka = k0 + half * 2;
        v2f a = *reinterpret_cast<const v2f*>(xrow + ka);   // 8B aligned (K even)
        const float* wr0 = Wm + (size_t)ka * DO + tn0 + m;
        const float* wr1 = wr0 + DO;
#pragma unroll
        for (int t = 0; t < NT; ++t) {
            v2f b;
            b.x = wr0[t * 16];
            b.y = wr1[t * 16];
            acc[t] = __builtin_amdgcn_wmma_f32_16x16x4_f32(false, a, false, b,
                                                           (short)0, acc[t],
                                                           false, false);
        }
    }
    const int row0 = tm + half * 8;
#pragma unroll
    for (int t = 0; t < NT; ++t) {
        float* hp = H + (size_t)row0 * DO + tn0 + t * 16 + m;
#pragma unroll
        for (int r = 0; r < 8; ++r) hp[(size_t)r * DO] = acc[t][r];
    }
}

// ---------------------------------------------------------------------------
// per-node attention scores: ssrc[n] = h[n].asrc, sdst[n] = h[n].adst
// one wave32 per node, shuffle reduction
// ---------------------------------------------------------------------------
template<int D>
__global__ __launch_bounds__(256) void scores_kernel(const float* __restrict__ H,
                                                     const float* __restrict__ asrc,
                                                     const float* __restrict__ adst,
                                                     float* __restrict__ ssrc,
                                                     float* __restrict__ sdst, int n)
{
    const int node = blockIdx.x * 8 + (threadIdx.x >> 5);
    const int lane = threadIdx.x & 31;
    if (node >= n) return;
    const float* row = H + (size_t)node * D;
    float s1 = 0.f, s2 = 0.f;
#pragma unroll
    for (int c = lane; c < D; c += 32) {
        float h = row[c];
        s1 += h * asrc[c];
        s2 += h * adst[c];
    }
#pragma unroll
    for (int off = 16; off; off >>= 1) {
        s1 += __shfl_xor(s1, off, 32);
        s2 += __shfl_xor(s2, off, 32);
    }
    if (lane == 0) { ssrc[node] = s1; sdst[node] = s2; }
}

// ---------------------------------------------------------------------------
// helpers
// ---------------------------------------------------------------------------
__device__ __forceinline__ void atomicMaxF32(float* addr, float v)
{
    // order-preserving int punning (works with -inf init = 0xFF800000)
    if (v >= 0.f) atomicMax((int*)addr, __float_as_int(v));
    else          atomicMin((unsigned int*)addr, __float_as_uint(v));
}

__global__ void fill_kernel(float* __restrict__ p, float v, int n)
{
    int i = blockIdx.x * blockDim.x + threadIdx.x;
    if (i < n) p[i] = v;
}

__global__ void bias_rows_kernel(float* __restrict__ out, const float* __restrict__ b,
                                 int n, int D)
{
    int i = blockIdx.x * blockDim.x + threadIdx.x;
    if (i < n * D) out[i] = b[i % D];
}

// ---------------------------------------------------------------------------
// edge passes (edge ids >= E are the implicit self-loops)
// ---------------------------------------------------------------------------
__global__ void edge_max_kernel(const int* __restrict__ src, const int* __restrict__ dst,
                                const float* __restrict__ ssrc, const float* __restrict__ sdst,
                                float* __restrict__ mmax, int E, int total)
{
    int e = blockIdx.x * blockDim.x + threadIdx.x;
    if (e >= total) return;
    int s = e < E ? src[e] : e - E;
    int d = e < E ? dst[e] : e - E;
    float v = ssrc[s] + sdst[d];
    v = v > 0.f ? v : NEG_SLOPE * v;
    atomicMaxF32(&mmax[d], v);
}

__global__ void edge_expsum_kernel(const int* __restrict__ src, const int* __restrict__ dst,
                                   const float* __restrict__ ssrc, const float* __restrict__ sdst,
                                   const float* __restrict__ mmax, float* __restrict__ denom,
                                   float* __restrict__ ex, int E, int total)
{
    int e = blockIdx.x * blockDim.x + threadIdx.x;
    if (e >= total) return;
    int s = e < E ? src[e] : e - E;
    int d = e < E ? dst[e] : e - E;
    float v = ssrc[s] + sdst[d];
    v = v > 0.f ? v : NEG_SLOPE * v;
    float xv = expf(v - mmax[d]);
    ex[e] = xv;
    atomicAdd(&denom[d], xv);
}

// each thread owns 4 channels of one edge: one b128 gather + 4 f32 atomics
template<int D>
__global__ __launch_bounds__(256) void edge_scatter_kernel(const int* __restrict__ src,
                                                           const int* __restrict__ dst,
                                                           const float* __restrict__ H,
                                                           const float* __restrict__ ex,
                                                           const float* __restrict__ denom,
                                                           float* __restrict__ OUT,
                                                           int E, int total)
{
    constexpr int TPE = D / 4;                 // threads per edge (16/32/64)
    constexpr int EPB = 256 / TPE;             // edges per block
    const int le = threadIdx.x / TPE;
    const int c4 = (threadIdx.x % TPE) * 4;
    const int e  = blockIdx.x * EPB + le;
    if (e >= total) return;
    int s = e < E ? src[e] : e - E;
    int d = e < E ? dst[e] : e - E;
    float alpha = ex[e] / denom[d];
    const float4 hv = *reinterpret_cast<const float4*>(H + (size_t)s * D + c4);
    float* o = OUT + (size_t)d * D + c4;
    atomicAdd(o + 0, hv.x * alpha);
    atomicAdd(o + 1, hv.y * alpha);
    atomicAdd(o + 2, hv.z * alpha);
    atomicAdd(o + 3, hv.w * alpha);
}

// ---------------------------------------------------------------------------
// pooling (h3 has D=256)
// ---------------------------------------------------------------------------
__global__ __launch_bounds__(256) void pool_kernel(const float* __restrict__ H,
                                                   const int* __restrict__ batch,
                                                   float* __restrict__ x1,
                                                   float* __restrict__ x2)
{
    int node = blockIdx.x;
    int c    = threadIdx.x;
    int g    = batch[node];
    float h  = H[(size_t)node * 256 + c];
    atomicMaxF32(&x1[g * 256 + c], h);
    atomicAdd(&x2[g * 256 + c], h);
}

__global__ void count_kernel(const int* __restrict__ batch, float* __restrict__ cnt, int n)
{
    int i = blockIdx.x * blockDim.x + threadIdx.x;
    if (i < n) atomicAdd(&cnt[batch[i]], 1.0f);
}

__global__ void meanpool_kernel(const float* __restrict__ x2, const float* __restrict__ cnt,
                                float* __restrict__ x3)
{
    int i = blockIdx.x * blockDim.x + threadIdx.x;
    if (i < G_GRAPHS * 256) x3[i] = x2[i] / fmaxf(cnt[i >> 8], 1.0f);
}

__global__ void concat_kernel(const float* __restrict__ x1, const float* __restrict__ x2,
                              float* __restrict__ z0)
{
    int i = blockIdx.x * blockDim.x + threadIdx.x;
    if (i < G_GRAPHS * 512) {
        int g = i >> 9, c = i & 511;
        z0[i] = (c < 256) ? x1[g * 256 + c] : x2[g * 256 + (c - 256)];
    }
}

// act: 0=none 1=relu 2=sigmoid
__global__ void dense_kernel(const float* __restrict__ in, const float* __restrict__ w,
                             const float* __restrict__ b, float* __restrict__ out,
                             int Gn, int Di, int Do, int act)
{
    int i = blockIdx.x * blockDim.x + threadIdx.x;
    if (i >= Gn * Do) return;
    int g = i / Do, o = i % Do;
    const float* row = in + (size_t)g * Di;
    float acc = b[o];
    for (int k = 0; k < Di; ++k) acc += row[k] * w[(size_t)k * Do + o];
    if (act == 1)      acc = fmaxf(acc, 0.f);
    else if (act == 2) acc = 1.f / (1.f + expf(-acc));
    out[i] = acc;
}

__global__ void mul_kernel(float* __restrict__ a, const float* __restrict__ b, int n)
{
    int i = blockIdx.x * blockDim.x + threadIdx.x;
    if (i < n) a[i] *= b[i];
}

// ---------------------------------------------------------------------------
// one full GAT layer
// ---------------------------------------------------------------------------
template<int K, int DO, int NT>
static void gat_layer(const float* Xin, const float* Wm, const float* asrc,
                      const float* adst, const float* bias,
                      const int* src, const int* dst,
                      float* H, float* OUT, float* ssrc, float* sdst,
                      float* mmax, float* denom, float* ex, hipStream_t stream)
{
    const int n = N_NODES;
    {
        constexpr int NSTRIP = DO / (16 * NT);
        int strips = (n / 16) * NSTRIP;
        gemm_wmma_f32<K, DO, NT><<<(strips + 7) / 8, 256, 0, stream>>>(Xin, Wm, H, n);
    }
    scores_kernel<DO><<<(n + 7) / 8, 256, 0, stream>>>(H, asrc, adst, ssrc, sdst, n);
    fill_kernel<<<(n + 255) / 256, 256, 0, stream>>>(mmax, -INFINITY, n);
    fill_kernel<<<(n + 255) / 256, 256, 0, stream>>>(denom, 0.0f, n);
    edge_max_kernel<<<(E_TOTAL + 255) / 256, 256, 0, stream>>>(src, dst, ssrc, sdst,
                                                               mmax, E_EDGES, E_TOTAL);
    edge_expsum_kernel<<<(E_TOTAL + 255) / 256, 256, 0, stream>>>(src, dst, ssrc, sdst,
                                                                  mmax, denom, ex,
                                                                  E_EDGES, E_TOTAL);
    bias_rows_kernel<<<(n * DO + 255) / 256, 256, 0, stream>>>(OUT, bias, n, DO);
    constexpr int EPB = 256 / (DO / 4);
    edge_scatter_kernel<DO><<<(E_TOTAL + EPB - 1) / EPB, 256, 0, stream>>>(
        src, dst, H, ex, denom, OUT, E_EDGES, E_TOTAL);
}

// ---------------------------------------------------------------------------
extern "C" void kernel_launch(void* const* d_in, const int* in_sizes, int n_in,
                              void* d_out, int out_size, void* d_ws, size_t ws_size,
                              hipStream_t stream)
{
    (void)in_sizes; (void)n_in; (void)out_size; (void)ws_size;

    const float* x     = (const float*)d_in[0];
    const int*   eidx  = (const int*)d_in[1];      // [2, E]
    const int*   batch = (const int*)d_in[2];
    const int*   src   = eidx;
    const int*   dst   = eidx + E_EDGES;

    const float* W1 = (const float*)d_in[3];
    const float* a1s = (const float*)d_in[4];
    const float* a1d = (const float*)d_in[5];
    const float* b1  = (const float*)d_in[6];
    const float* W2 = (const float*)d_in[7];
    const float* a2s = (const float*)d_in[8];
    const float* a2d = (const float*)d_in[9];
    const float* b2  = (const float*)d_in[10];
    const float* W3 = (const float*)d_in[11];
    const float* a3s = (const float*)d_in[12];
    const float* a3d = (const float*)d_in[13];
    const float* b3  = (const float*)d_in[14];
    const float* d1w = (const float*)d_in[15]; const float* d1b = (const float*)d_in[16];
    const float* d2w = (const float*)d_in[17]; const float* d2b = (const float*)d_in[18];
    const float* d3w = (const float*)d_in[19]; const float* d3b = (const float*)d_in[20];
    const float* mw  = (const float*)d_in[21]; const float* mb  = (const float*)d_in[22];
    const float* d4w = (const float*)d_in[23]; const float* d4b = (const float*)d_in[24];
    const float* d5w = (const float*)d_in[25]; const float* d5b = (const float*)d_in[26];
    const float* d6w = (const float*)d_in[27]; const float* d6b = (const float*)d_in[28];
    const float* d7w = (const float*)d_in[29]; const float* d7b = (const float*)d_in[30];
    float* out = (float*)d_out;

    // workspace bump allocator (floats)
    float* ws = (float*)d_ws;
    size_t off = 0;
    auto alloc = [&](size_t n) { float* p = ws + off; off += n; return p; };

    float* A     = alloc((size_t)N_NODES * 256);   // transform result h
    float* B     = alloc((size_t)N_NODES * 256);   // aggregated output / next input
    float* ssrc  = alloc(N_NODES);
    float* sdst  = alloc(N_NODES);
    float* mmax  = alloc(N_NODES);
    float* denom = alloc(N_NODES);
    float* ex    = alloc(E_TOTAL);
    float* x1    = alloc(G_GRAPHS * 256);
    float* x2    = alloc(G_GRAPHS * 256);
    float* cnt   = alloc(G_GRAPHS);
    float* x3    = alloc(G_GRAPHS * 256);
    float* z0    = alloc(G_GRAPHS * 512);
    float* z1    = alloc(G_GRAPHS * 256);
    float* z2    = alloc(G_GRAPHS * 128);
    float* z3    = alloc(G_GRAPHS * 64);
    float* gate  = alloc(G_GRAPHS * 64);
    float* z4    = alloc(G_GRAPHS * 64);
    float* z5    = alloc(G_GRAPHS * 128);
    float* z6    = alloc(G_GRAPHS * 256);

    // --- three GAT layers (ping-pong A/B; B becomes next layer input) ---
    // NT: column tiles per wave strip (A-fragment reuse factor)
    gat_layer<128,  64, 4>(x, W1, a1s, a1d, b1, src, dst, A, B, ssrc, sdst, mmax, denom, ex, stream);
    gat_layer< 64, 128, 8>(B, W2, a2s, a2d, b2, src, dst, A, B, ssrc, sdst, mmax, denom, ex, stream);
    gat_layer<128, 256, 8>(B, W3, a3s, a3d, b3, src, dst, A, B, ssrc, sdst, mmax, denom, ex, stream);
    // h3 is now in B [N, 256]

    // --- pooling ---
    fill_kernel<<<(G_GRAPHS * 256 + 255) / 256, 256, 0, stream>>>(x1, -INFINITY, G_GRAPHS * 256);
    fill_kernel<<<(G_GRAPHS * 256 + 255) / 256, 256, 0, stream>>>(x2, 0.0f, G_GRAPHS * 256);
    fill_kernel<<<1, 256, 0, stream>>>(cnt, 0.0f, G_GRAPHS);
    pool_kernel<<<N_NODES, 256, 0, stream>>>(B, batch, x1, x2);
    count_kernel<<<(N_NODES + 255) / 256, 256, 0, stream>>>(batch, cnt, N_NODES);
    meanpool_kernel<<<(G_GRAPHS * 256 + 255) / 256, 256, 0, stream>>>(x2, cnt, x3);
    concat_kernel<<<(G_GRAPHS * 512 + 255) / 256, 256, 0, stream>>>(x1, x2, z0);

    // --- dense tail ---
    auto dense = [&](const float* in_, const float* w_, const float* b_, float* out_,
                     int Di, int Do, int act) {
        dense_kernel<<<(G_GRAPHS * Do + 255) / 256, 256, 0, stream>>>(in_, w_, b_, out_,
                                                                      G_GRAPHS, Di, Do, act);
    };
    dense(z0, d1w, d1b, z1, 512, 256, 1);
    dense(z1, d2w, d2b, z2, 256, 128, 1);
    dense(z2, d3w, d3b, z3, 128,  64, 1);
    dense(x3, mw,  mb,  gate, 256, 64, 2);
    mul_kernel<<<(G_GRAPHS * 64 + 255) / 256, 256, 0, stream>>>(z3, gate, G_GRAPHS * 64);
    dense(z3, d4w, d4b, z4,  64,  64, 1);
    dense(z4, d5w, d5b, z5,  64, 128, 1);
    dense(z5, d6w, d6b, z6, 128, 256, 1);
    dense(z6, d7w, d7b, out, 256, 128, 0);
}